// VolumetricRenderer_67156108640756
// MI455X (gfx1250) — compile-verified
//
#include <hip/hip_runtime.h>
#include <stdint.h>

// NeRF raw2outputs for MI455X (gfx1250, wave32).
// One wave = one ray; 6 chunks of 32 samples; wave-level product scan for the
// exclusive transmittance cumprod; async global->LDS double buffering for the
// dominant raw[] stream via gfx1250 global_load_async_to_lds_b128.

#define NUM_SAMPLES 192
#define CHUNK 32
#define NCHUNK (NUM_SAMPLES / CHUNK) // 6
#define WAVES_PER_BLOCK 8
#define BLOCK_THREADS (WAVES_PER_BLOCK * 32)
#define INF_DIST 1e10f

#if defined(__has_builtin)
#if __has_builtin(__builtin_amdgcn_global_load_async_to_lds_b128) && \
    __has_builtin(__builtin_amdgcn_s_wait_asynccnt)
#define USE_ASYNC 1
#endif
#endif
#ifndef USE_ASYNC
#define USE_ASYNC 0
#endif

// Builtin param type per hipcc diagnostic:
//   __attribute__((__vector_size__(4*sizeof(int)))) int __device__ *   (AS1 int4*)
typedef int v4i __attribute__((__vector_size__(4 * sizeof(int))));
typedef __attribute__((address_space(1))) v4i as1_v4i;
typedef __attribute__((address_space(3))) v4i as3_v4i;

__device__ __forceinline__ float wave_sum(float v) {
#pragma unroll
  for (int off = 16; off > 0; off >>= 1) v += __shfl_xor(v, off, 32);
  return v;
}

__device__ __forceinline__ float sigmoidf_fast(float x) {
  return 1.0f / (1.0f + __expf(-x));
}

__global__ __launch_bounds__(BLOCK_THREADS)
void nerf_raw2outputs(const float4* __restrict__ raw,    // [N, S] as float4 (rgb,sigma)
                      const float*  __restrict__ z_vals, // [N, S]
                      const float*  __restrict__ rays_d, // [N, 3]
                      float* __restrict__ out, int N) {
  const int wave = (int)(threadIdx.x >> 5);
  const int lane = (int)(threadIdx.x & 31);
  const int ray  = (int)blockIdx.x * WAVES_PER_BLOCK + wave;
  if (ray >= N) return;

  // |rays_d| for this ray (broadcast loads; uniform across the wave)
  const float dx = rays_d[(size_t)ray * 3 + 0];
  const float dy = rays_d[(size_t)ray * 3 + 1];
  const float dz = rays_d[(size_t)ray * 3 + 2];
  const float nrm = sqrtf(dx * dx + dy * dy + dz * dz);

  const float4* rawRow = raw + (size_t)ray * NUM_SAMPLES;
  const float*  zRow   = z_vals + (size_t)ray * NUM_SAMPLES;
  float* wRow = out + (size_t)N * 5 + (size_t)ray * NUM_SAMPLES; // weights block

#if USE_ASYNC
  __shared__ float4 stage[2][WAVES_PER_BLOCK][CHUNK];
  // Kick off chunk 0: per-lane 16B async copy global -> LDS (ASYNCcnt path).
  __builtin_amdgcn_global_load_async_to_lds_b128(
      (as1_v4i*)(uintptr_t)(rawRow + lane),
      (as3_v4i*)(uint32_t)(uintptr_t)&stage[0][wave][lane], 0, 0);
#endif

  float Tcarry = 1.0f; // running exclusive transmittance product across chunks
  float accR = 0.0f, accG = 0.0f, accB = 0.0f, accD = 0.0f, accW = 0.0f;

#pragma unroll
  for (int c = 0; c < NCHUNK; ++c) {
    const int s = c * CHUNK + lane;

#if USE_ASYNC
    if (c + 1 < NCHUNK) {
      __builtin_amdgcn_global_load_async_to_lds_b128(
          (as1_v4i*)(uintptr_t)(rawRow + (c + 1) * CHUNK + lane),
          (as3_v4i*)(uint32_t)(uintptr_t)&stage[(c + 1) & 1][wave][lane], 0, 0);
      __builtin_amdgcn_s_wait_asynccnt(1); // oldest (chunk c) complete
    } else {
      __builtin_amdgcn_s_wait_asynccnt(0);
    }
    const float4 r4 = stage[c & 1][wave][lane];
#else
    const float4 r4 = rawRow[s]; // coalesced global_load_b128
#endif

    const float z = zRow[s];
    // dists = z[s+1] - z[s]; last interval = INF. Neighbor via lane shuffle,
    // chunk boundary patched by a single lane-31 load.
    float zn = __shfl_down(z, 1, 32);
    if (lane == 31 && s + 1 < NUM_SAMPLES) zn = zRow[s + 1];
    float dist = (s == NUM_SAMPLES - 1) ? INF_DIST : (zn - z);
    dist *= nrm;

    const float sigma = fmaxf(r4.w, 0.0f);
    const float alpha = 1.0f - __expf(-sigma * dist);
    float p = 1.0f - alpha + 1e-10f;

    // Inclusive product scan across the wave (Hillis-Steele, 5 steps).
#pragma unroll
    for (int off = 1; off < 32; off <<= 1) {
      const float v = __shfl_up(p, off, 32);
      if (lane >= off) p *= v;
    }
    float excl = __shfl_up(p, 1, 32);
    if (lane == 0) excl = 1.0f;

    const float T = Tcarry * excl;     // exclusive cumprod at sample s
    const float w = alpha * T;         // weight
    Tcarry *= __shfl(p, 31, 32);       // carry full-chunk product forward

    accR += w * sigmoidf_fast(r4.x);
    accG += w * sigmoidf_fast(r4.y);
    accB += w * sigmoidf_fast(r4.z);
    accD += w * z;
    accW += w;

    wRow[s] = w; // coalesced b32 store of weights
  }

  const float r   = wave_sum(accR);
  const float g   = wave_sum(accG);
  const float b   = wave_sum(accB);
  const float dep = wave_sum(accD);
  const float acc = wave_sum(accW);

  if (lane == 0) {
    const float depth = (acc <= 1e-10f) ? INF_DIST : dep;
    const float disp  = 1.0f / fmaxf(1e-10f, depth / acc);
    // Output layout: rgb[3N] | disp[N] | acc[N] | weights[192N] | depth[N]
    out[(size_t)ray * 3 + 0] = r;
    out[(size_t)ray * 3 + 1] = g;
    out[(size_t)ray * 3 + 2] = b;
    out[(size_t)N * 3 + ray] = disp;
    out[(size_t)N * 4 + ray] = acc;
    out[(size_t)N * 5 + (size_t)N * NUM_SAMPLES + ray] = depth;
  }
}

extern "C" void kernel_launch(void* const* d_in, const int* in_sizes, int n_in,
                              void* d_out, int out_size, void* d_ws, size_t ws_size,
                              hipStream_t stream) {
  (void)n_in; (void)out_size; (void)d_ws; (void)ws_size;
  const float4* raw    = (const float4*)d_in[0];
  const float*  z_vals = (const float*)d_in[1];
  const float*  rays_d = (const float*)d_in[2];
  float* out = (float*)d_out;

  const int N = in_sizes[1] / NUM_SAMPLES; // z_vals is [N, S]
  const int blocks = (N + WAVES_PER_BLOCK - 1) / WAVES_PER_BLOCK;
  hipLaunchKernelGGL(nerf_raw2outputs, dim3(blocks), dim3(BLOCK_THREADS), 0,
                     stream, raw, z_vals, rays_d, out, N);
}